// NonLinearBodyOrderedModel_6184752906370
// MI455X (gfx1250) — compile-verified
//
#include <hip/hip_runtime.h>

typedef __attribute__((ext_vector_type(2))) float v2f;
typedef __attribute__((ext_vector_type(8))) float v8f;

#define NLM 16
#define RMAX 5.0f
#define INV_AVG 0.0625f
#define SQRT4PI 3.5449077018110318f
#define PI_F 3.14159265358979323846f
#define CBESS 0.6324555320336759f /* sqrt(2/5) */

__device__ __forceinline__ v8f wmma_f32_4(v2f a, v2f b, v8f c) {
  return __builtin_amdgcn_wmma_f32_16x16x4_f32(false, a, false, b, (short)0, c,
                                               false, false);
}
__device__ __forceinline__ int l_of_lm(int lm) {
  return (lm == 0) ? 0 : (lm < 4) ? 1 : (lm < 9) ? 2 : 3;
}

// ---------------------------------------------------------------- zero
__global__ void k_zero(float* p, long n) {
  long i = (long)blockIdx.x * blockDim.x + threadIdx.x;
  if (i < n) p[i] = 0.f;
}

// ---------------------------------------------------------------- node embed + e0
__global__ void k_embed(const float* __restrict__ na, const float* __restrict__ We,
                        const float* __restrict__ ae, const int* __restrict__ batch,
                        float* __restrict__ h, float* __restrict__ e0g, int N) {
  int tid = blockIdx.x * blockDim.x + threadIdx.x;
  int n = tid >> 6, k = tid & 63;
  if (n >= N) return;
  float acc = 0.f;
#pragma unroll
  for (int s = 0; s < 10; ++s) acc += na[n * 10 + s] * We[s * 64 + k];
  h[n * 64 + k] = acc;
  if (k == 0) {
    float e0 = 0.f;
#pragma unroll
    for (int s = 0; s < 10; ++s) e0 += na[n * 10 + s] * ae[s];
    atomicAdd(&e0g[batch[n]], e0);
  }
}

// ---------------------------------------------------------------- edge geometry
__device__ __forceinline__ void edge_geom(float vx, float vy, float vz, bool grad,
                                          float& r, float& inv_r,
                                          float& x, float& y, float& z,
                                          float* Y, float* ef, float* defdr) {
  r = sqrtf(vx * vx + vy * vy + vz * vz + 1e-12f);
  inv_r = 1.f / r;
  x = vx * inv_r; y = vy * inv_r; z = vz * inv_r;
  float x2 = x * x, y2 = y * y, z2 = z * z;
  const float c1 = 0.4886025119029199f, c2 = 1.0925484305920792f;
  const float c3 = 0.31539156525252005f, c4 = 0.5462742152960396f;
  const float c5 = 0.5900435899266435f, c6 = 2.890611442640554f;
  const float c7 = 0.4570457994644658f, c8 = 0.3731763325901154f;
  const float c9 = 1.445305721320277f;
  Y[0] = 0.28209479177387814f * SQRT4PI;
  Y[1] = c1 * y * SQRT4PI;                 Y[2] = c1 * z * SQRT4PI;
  Y[3] = c1 * x * SQRT4PI;                 Y[4] = c2 * x * y * SQRT4PI;
  Y[5] = c2 * y * z * SQRT4PI;             Y[6] = c3 * (3.f * z2 - 1.f) * SQRT4PI;
  Y[7] = c2 * x * z * SQRT4PI;             Y[8] = c4 * (x2 - y2) * SQRT4PI;
  Y[9] = c5 * y * (3.f * x2 - y2) * SQRT4PI;
  Y[10] = c6 * x * y * z * SQRT4PI;
  Y[11] = c7 * y * (5.f * z2 - 1.f) * SQRT4PI;
  Y[12] = c8 * z * (5.f * z2 - 3.f) * SQRT4PI;
  Y[13] = c7 * x * (5.f * z2 - 1.f) * SQRT4PI;
  Y[14] = c9 * z * (x2 - y2) * SQRT4PI;
  Y[15] = c5 * x * (x2 - 3.f * y2) * SQRT4PI;
  // cutoff (p=6): 1 - 28 x^6 + 48 x^7 - 21 x^8
  float xr = r * (1.f / RMAX);
  float fc = 0.f, dfc = 0.f;
  if (xr < 1.f) {
    float xr2 = xr * xr, xr3 = xr2 * xr, xr5 = xr3 * xr2, xr6 = xr3 * xr3;
    fc = 1.f - 28.f * xr6 + 48.f * xr6 * xr - 21.f * xr6 * xr2;
    dfc = (-168.f * xr5 + 336.f * xr6 - 168.f * xr6 * xr) * (1.f / RMAX);
  }
#pragma unroll
  for (int b = 0; b < 8; ++b) {
    float a = (float)(b + 1) * (PI_F / RMAX);
    float s, c;
    __sincosf(a * r, &s, &c);
    float bess = CBESS * s * inv_r;
    ef[b] = bess * fc;
    if (grad) {
      float dbess = CBESS * inv_r * (a * c - s * inv_r);
      defdr[b] = dbess * fc + bess * dfc;
    }
  }
}

// ---------------------------------------------------------------- edge forward
// A layout: [n][lm][k]  (A[n*1024 + lm*64 + k]) -> coalesced lane access (k = lane)
__global__ void k_edge_fwd(const float* __restrict__ pos, const float* __restrict__ shifts,
                           const int* __restrict__ eidx, const float* __restrict__ h,
                           const float* __restrict__ Wr1, const float* __restrict__ Wr2,
                           float* __restrict__ A, int E) {
  int e = blockIdx.x * 8 + (threadIdx.x >> 5);
  int lane = threadIdx.x & 31;
  if (e >= E) return;
  int send = eidx[e], recv = eidx[E + e];
  float vx = pos[recv * 3 + 0] - pos[send * 3 + 0] + shifts[e * 3 + 0];
  float vy = pos[recv * 3 + 1] - pos[send * 3 + 1] + shifts[e * 3 + 1];
  float vz = pos[recv * 3 + 2] - pos[send * 3 + 2] + shifts[e * 3 + 2];
  float r, inv_r, x, y, z, Y[NLM], ef[8], dummy[8];
  edge_geom(vx, vy, vz, false, r, inv_r, x, y, z, Y, ef, dummy);
  // radial MLP: 8 -> 64 (silu) -> 64
  float pre0 = 0.f, pre1 = 0.f;
#pragma unroll
  for (int b = 0; b < 8; ++b) {
    pre0 += ef[b] * Wr1[b * 64 + lane];
    pre1 += ef[b] * Wr1[b * 64 + lane + 32];
  }
  float sg0 = 1.f / (1.f + __expf(-pre0)), sg1 = 1.f / (1.f + __expf(-pre1));
  float sl0 = pre0 * sg0, sl1 = pre1 * sg1;
  float w0 = 0.f, w1 = 0.f;
  for (int j = 0; j < 32; ++j) {
    float a = __shfl(sl0, j, 32), b = __shfl(sl1, j, 32);
    w0 += a * Wr2[j * 64 + lane] + b * Wr2[(j + 32) * 64 + lane];
    w1 += a * Wr2[j * 64 + lane + 32] + b * Wr2[(j + 32) * 64 + lane + 32];
  }
  float t0 = w0 * h[send * 64 + lane] * INV_AVG;
  float t1 = w1 * h[send * 64 + lane + 32] * INV_AVG;
  float* Ar = A + (long)recv * 1024;
#pragma unroll
  for (int lm = 0; lm < NLM; ++lm) {
    atomicAdd(Ar + lm * 64 + lane, t0 * Y[lm]);
    atomicAdd(Ar + lm * 64 + lane + 32, t1 * Y[lm]);
  }
}

// ---------------------------------------------------------------- node kernel (WMMA)
// Per block: 16 nodes, 256 threads (8 wave32).
//  phase0: stage 64KB A tile into LDS via GLOBAL_LOAD_ASYNC_TO_LDS_B128 (ASYNCcnt)
//  phase1: Am[lm][n][c] = sum_k A[n][lm][k] * Wmix[l][k][c]  (WMMA f32 16x16x4,
//          A-fragments from LDS, accumulators held in registers)
//  phase1b: write Am tiles into LDS (reusing the staging buffer)
//  phase2: s, a0, B, node_e -> e1; transform Am -> dAm in LDS
//  phase3: dA[n][lm][k] = sum_c dAm[n][c][lm] * Wmix[l][k][c] (WMMA f32 16x16x4)
__global__ void __launch_bounds__(256)
k_node(const float* __restrict__ A, const float* __restrict__ Wmix,
       const float* __restrict__ Wp, const float* __restrict__ wro,
       const int* __restrict__ batch, float* __restrict__ dA,
       float* __restrict__ e1g, int N) {
  __shared__ float Ams[16 * 1024];  // 64 KB: A-tile, then Am/dAm
  int n0 = blockIdx.x * 16;
  int tid = threadIdx.x;
  int wave = tid >> 5, lane = tid & 31;
  int mrow = lane & 15, half = lane >> 4;

  // ---- phase 0: async copy of A tile -> LDS
  {
    int rows = min(16, N - n0);
    int vbytes = rows * 4096;
    unsigned lds0 = (unsigned)(unsigned long long)(uintptr_t)&Ams[0];
    const char* gbase = (const char*)(A + (long)n0 * 1024);
    for (int ofs = tid * 16; ofs < vbytes; ofs += 256 * 16) {
      unsigned la = lds0 + (unsigned)ofs;
      asm volatile("global_load_async_to_lds_b128 %0, %1, %2"
                   :
                   : "v"(la), "v"(ofs), "s"(gbase)
                   : "memory");
    }
    asm volatile("s_wait_asynccnt 0" ::: "memory");
  }
  __syncthreads();

  // ---- phase 1: Am = A @ W (all 8 job tiles accumulated in registers)
  v8f acc[8];
#pragma unroll
  for (int j = 0; j < 8; ++j) {
    int job = wave * 8 + j;
    int lm = job >> 2, c0 = (job & 3) << 4;
    const float* Wl = Wmix + l_of_lm(lm) * 4096;
    const float* Ap = &Ams[mrow * 1024 + lm * 64];  // LDS, contiguous k
    v8f a8 = {0.f, 0.f, 0.f, 0.f, 0.f, 0.f, 0.f, 0.f};
    acc[j] = a8;
    for (int kk = 0; kk < 16; ++kk) {
      int k0 = kk * 4 + half * 2;
      v2f a = *(const v2f*)(Ap + k0);  // ds_load_b64
      v2f b;
      b.x = Wl[k0 * 64 + c0 + mrow];
      b.y = Wl[(k0 + 1) * 64 + c0 + mrow];
      acc[j] = wmma_f32_4(a, b, acc[j]);
    }
  }
  __syncthreads();  // all tile reads complete before overwrite

  // ---- phase 1b: store Am tiles into LDS as [lm][n][c]
#pragma unroll
  for (int j = 0; j < 8; ++j) {
    int job = wave * 8 + j;
    int lm = job >> 2, c0 = (job & 3) << 4;
#pragma unroll
    for (int v = 0; v < 8; ++v)
      Ams[lm * 1024 + (v + half * 8) * 64 + c0 + mrow] = acc[j][v];
  }
  __syncthreads();

  // ---- phase 2: nonlinearity, e1, Am -> dAm (in place)
  {
    int n = tid & 15;
    int cb = (tid >> 4) * 4;
    float pe = 0.f;
    for (int jj = 0; jj < 4; ++jj) {
      int c = cb + jj;
      float* col = &Ams[n * 64 + c];
      float a0 = col[0];
      float s = 0.f;
#pragma unroll
      for (int lm = 0; lm < NLM; ++lm) { float v = col[lm * 1024]; s += v * v; }
      float wr = wro[c];
      float gs = wr * (Wp[64 + c] + Wp[128 + c] * a0);  // dB/ds * w_ro
      float ga = wr * (Wp[c] + Wp[128 + c] * s);        // dB/da0 * w_ro
      pe += wr * (Wp[c] * a0 + Wp[64 + c] * s + Wp[128 + c] * a0 * s);
      float g2 = 2.f * gs;
#pragma unroll
      for (int lm = 0; lm < NLM; ++lm)
        col[lm * 1024] = g2 * col[lm * 1024] + ((lm == 0) ? ga : 0.f);
    }
    if (n0 + n < N) atomicAdd(&e1g[batch[n0 + n]], pe);
  }
  __syncthreads();

  // ---- phase 3: dA = dAm @ W^T  (coalesced stores, dA layout [n][lm][k])
  for (int j = 0; j < 8; ++j) {
    int job = wave * 8 + j;
    int lm = job >> 2, k0 = (job & 3) << 4;
    const float* Wl = Wmix + l_of_lm(lm) * 4096;
    v8f d = {0.f, 0.f, 0.f, 0.f, 0.f, 0.f, 0.f, 0.f};
    for (int cc = 0; cc < 16; ++cc) {
      int c0 = cc * 4 + half * 2;
      v2f a = *(const v2f*)(&Ams[lm * 1024 + mrow * 64 + c0]);  // ds_load_b64
      v2f b;
      b.x = Wl[(k0 + mrow) * 64 + c0];  // B[K=c][N=k] = W[k][c]
      b.y = Wl[(k0 + mrow) * 64 + c0 + 1];
      d = wmma_f32_4(a, b, d);
    }
#pragma unroll
    for (int v = 0; v < 8; ++v) {
      int n = n0 + v + half * 8;
      if (n < N) dA[(long)n * 1024 + lm * 64 + k0 + mrow] = d[v];
    }
  }
}

// ---------------------------------------------------------------- edge backward
__global__ void k_edge_bwd(const float* __restrict__ pos, const float* __restrict__ shifts,
                           const int* __restrict__ eidx, const float* __restrict__ h,
                           const float* __restrict__ Wr1, const float* __restrict__ Wr2,
                           const float* __restrict__ dA, float* __restrict__ forces, int E) {
  int e = blockIdx.x * 8 + (threadIdx.x >> 5);
  int lane = threadIdx.x & 31;
  if (e >= E) return;
  int send = eidx[e], recv = eidx[E + e];
  const float* dAr = dA + (long)recv * 1024;
  __builtin_prefetch(dAr, 0, 1);  // global_prefetch
  float vx = pos[recv * 3 + 0] - pos[send * 3 + 0] + shifts[e * 3 + 0];
  float vy = pos[recv * 3 + 1] - pos[send * 3 + 1] + shifts[e * 3 + 1];
  float vz = pos[recv * 3 + 2] - pos[send * 3 + 2] + shifts[e * 3 + 2];
  float r, inv_r, x, y, z, Y[NLM], ef[8], defdr[8];
  edge_geom(vx, vy, vz, true, r, inv_r, x, y, z, Y, ef, defdr);
  // recompute MLP forward
  float pre0 = 0.f, pre1 = 0.f;
#pragma unroll
  for (int b = 0; b < 8; ++b) {
    pre0 += ef[b] * Wr1[b * 64 + lane];
    pre1 += ef[b] * Wr1[b * 64 + lane + 32];
  }
  float sg0 = 1.f / (1.f + __expf(-pre0)), sg1 = 1.f / (1.f + __expf(-pre1));
  float sl0 = pre0 * sg0, sl1 = pre1 * sg1;
  float w0 = 0.f, w1 = 0.f;
  for (int j = 0; j < 32; ++j) {
    float a = __shfl(sl0, j, 32), b = __shfl(sl1, j, 32);
    w0 += a * Wr2[j * 64 + lane] + b * Wr2[(j + 32) * 64 + lane];
    w1 += a * Wr2[j * 64 + lane + 32] + b * Wr2[(j + 32) * 64 + lane + 32];
  }
  float hk0 = h[send * 64 + lane], hk1 = h[send * 64 + lane + 32];
  float t0 = w0 * hk0 * INV_AVG, t1 = w1 * hk1 * INV_AVG;
  // gather dA[recv] (coalesced: k = lane); dY[lm] = sum_k t_k dA_k,lm
  float dY[NLM];
  float sum0 = 0.f, sum1 = 0.f;
#pragma unroll
  for (int lm = 0; lm < NLM; ++lm) {
    float d0 = dAr[lm * 64 + lane];
    float d1 = dAr[lm * 64 + lane + 32];
    dY[lm] = t0 * d0 + t1 * d1;
    sum0 += d0 * Y[lm];
    sum1 += d1 * Y[lm];
  }
#pragma unroll
  for (int lm = 0; lm < NLM; ++lm)
    for (int m = 16; m >= 1; m >>= 1) dY[lm] += __shfl_xor(dY[lm], m, 32);
  // backward through MLP -> d(edge_feats) -> f_r
  float dw0 = hk0 * INV_AVG * sum0, dw1 = hk1 * INV_AVG * sum1;
  float dsl0 = 0.f, dsl1 = 0.f;
  for (int k = 0; k < 32; ++k) {
    float a = __shfl(dw0, k, 32), b = __shfl(dw1, k, 32);
    dsl0 += a * Wr2[lane * 64 + k] + b * Wr2[lane * 64 + k + 32];
    dsl1 += a * Wr2[(lane + 32) * 64 + k] + b * Wr2[(lane + 32) * 64 + k + 32];
  }
  float dp0 = dsl0 * sg0 * (1.f + pre0 * (1.f - sg0));
  float dp1 = dsl1 * sg1 * (1.f + pre1 * (1.f - sg1));
  float fr = 0.f;
#pragma unroll
  for (int b = 0; b < 8; ++b) {
    float def = dp0 * Wr1[b * 64 + lane] + dp1 * Wr1[b * 64 + lane + 32];
    for (int m = 16; m >= 1; m >>= 1) def += __shfl_xor(def, m, 32);
    fr += def * defdr[b];
  }
  // dY -> du (analytic grads of real spherical harmonics), then dvec
  const float c1 = 0.4886025119029199f, c2 = 1.0925484305920792f;
  const float c3 = 0.31539156525252005f, c4 = 0.5462742152960396f;
  const float c5 = 0.5900435899266435f, c6 = 2.890611442640554f;
  const float c7 = 0.4570457994644658f, c8 = 0.3731763325901154f;
  const float c9 = 1.445305721320277f;
  float x2 = x * x, y2 = y * y, z2 = z * z;
  float dux = c1 * dY[3] + c2 * y * dY[4] + c2 * z * dY[7] + 2.f * c4 * x * dY[8]
            + 6.f * c5 * x * y * dY[9] + c6 * y * z * dY[10]
            + c7 * (5.f * z2 - 1.f) * dY[13] + 2.f * c9 * x * z * dY[14]
            + 3.f * c5 * (x2 - y2) * dY[15];
  float duy = c1 * dY[1] + c2 * x * dY[4] + c2 * z * dY[5] - 2.f * c4 * y * dY[8]
            + 3.f * c5 * (x2 - y2) * dY[9] + c6 * x * z * dY[10]
            + c7 * (5.f * z2 - 1.f) * dY[11] - 2.f * c9 * y * z * dY[14]
            - 6.f * c5 * x * y * dY[15];
  float duz = c1 * dY[2] + c2 * y * dY[5] + 6.f * c3 * z * dY[6] + c2 * x * dY[7]
            + c6 * x * y * dY[10] + 10.f * c7 * y * z * dY[11]
            + c8 * (15.f * z2 - 3.f) * dY[12] + 10.f * c7 * x * z * dY[13]
            + c9 * (x2 - y2) * dY[14];
  dux *= SQRT4PI; duy *= SQRT4PI; duz *= SQRT4PI;
  float udot = x * dux + y * duy + z * duz;
  float gvx = x * fr + (dux - x * udot) * inv_r;
  float gvy = y * fr + (duy - y * udot) * inv_r;
  float gvz = z * fr + (duz - z * udot) * inv_r;
  if (lane == 0) {
    // forces = -dE/dpos; vec = pos[recv]-pos[send]
    atomicAdd(&forces[send * 3 + 0], gvx);
    atomicAdd(&forces[send * 3 + 1], gvy);
    atomicAdd(&forces[send * 3 + 2], gvz);
    atomicAdd(&forces[recv * 3 + 0], -gvx);
    atomicAdd(&forces[recv * 3 + 1], -gvy);
    atomicAdd(&forces[recv * 3 + 2], -gvz);
  }
}

// ---------------------------------------------------------------- finalize
__global__ void k_final(const float* __restrict__ e0g, const float* __restrict__ e1g,
                        float* __restrict__ out, int G) {
  int g = threadIdx.x;
  if (g < G) {
    float a = e0g[g], b = e1g[g];
    out[g] = a + b;
    out[G + 2 * g] = a;
    out[G + 2 * g + 1] = b;
  }
}

// ================================================================ launch
extern "C" void kernel_launch(void* const* d_in, const int* in_sizes, int n_in,
                              void* d_out, int out_size, void* d_ws, size_t ws_size,
                              hipStream_t stream) {
  const float* pos    = (const float*)d_in[0];
  const float* na     = (const float*)d_in[1];
  const float* shifts = (const float*)d_in[2];
  const float* We     = (const float*)d_in[3];
  const float* ae     = (const float*)d_in[4];
  const float* Wr1    = (const float*)d_in[5];
  const float* Wr2    = (const float*)d_in[6];
  const float* Wmix   = (const float*)d_in[7];
  const float* Wp     = (const float*)d_in[8];
  const float* wro    = (const float*)d_in[9];
  const int*   eidx   = (const int*)d_in[10];
  const int*   batch  = (const int*)d_in[11];

  int N = in_sizes[0] / 3;
  int E = in_sizes[10] / 2;
  int G = (out_size - 3 * N) / 3;

  float* ws = (float*)d_ws;
  long offA  = 0;
  long offDA = offA + (long)N * 1024;
  long offH  = offDA + (long)N * 1024;
  long offE0 = offH + (long)N * 64;
  long offE1 = offE0 + G;
  long wtot  = offE1 + G;
  float* A   = ws + offA;
  float* dAb = ws + offDA;
  float* h   = ws + offH;
  float* e0g = ws + offE0;
  float* e1g = ws + offE1;
  float* out = (float*)d_out;
  float* forces = out + 3 * G;
  (void)n_in; (void)ws_size;

  k_zero<<<(int)((wtot + 255) / 256), 256, 0, stream>>>(ws, wtot);
  k_zero<<<(out_size + 255) / 256, 256, 0, stream>>>(out, out_size);
  k_embed<<<(N * 64 + 255) / 256, 256, 0, stream>>>(na, We, ae, batch, h, e0g, N);
  k_edge_fwd<<<(E + 7) / 8, 256, 0, stream>>>(pos, shifts, eidx, h, Wr1, Wr2, A, E);
  k_node<<<(N + 15) / 16, 256, 0, stream>>>(A, Wmix, Wp, wro, batch, dAb, e1g, N);
  k_edge_bwd<<<(E + 7) / 8, 256, 0, stream>>>(pos, shifts, eidx, h, Wr1, Wr2, dAb,
                                              forces, E);
  k_final<<<1, 256, 0, stream>>>(e0g, e1g, out, G);
}